// Rotation_14791867367981
// MI455X (gfx1250) — compile-verified
//
#include <hip/hip_runtime.h>
#include <cmath>
#include <complex>

// ---------------------------------------------------------------------------
// MI455X (gfx1250) rotation-of-irreps kernel.
//
// out[b] (16x256) = Dfull[b] (16x16 block-diag: [1 | D1(3x3) | D2(5x5) | D3(7x7)])
//                   @ x[b] (16x256)
//
// Memory bound: 268 MB @ 23.3 TB/s ~= 11.5 us floor. WMMA f32 16x16x4 does the
// apply; per-batch D matrices are built in LDS from host-precomputed
// eigendecomposition constants of the x-axis angular momentum generator.
// ---------------------------------------------------------------------------

typedef __attribute__((ext_vector_type(2))) float v2f;
typedef __attribute__((ext_vector_type(8))) float v8f;

struct RotConsts {
  float w1[3];  float Vr1[9];  float Vi1[9];
  float w2[5];  float Vr2[25]; float Vi2[25];
  float w3[7];  float Vr3[49]; float Vi3[49];
};

__global__ __launch_bounds__(256) void rot_irreps_wmma(
    const float* __restrict__ gamma,
    const float* __restrict__ beta,
    const float* __restrict__ alpha,
    const float* __restrict__ x,
    float* __restrict__ out,
    RotConsts K)
{
  __shared__ float Dm[16][17];   // padded: lanes 0-15 read a column -> no bank conflicts
  const int b = blockIdx.x;
  const int t = threadIdx.x;
  const float al = alpha[b];
  const float be = beta[b];
  const float ga = gamma[b];

  // ---------------- Phase 1: build Dfull (16x16) in LDS ---------------------
  // Row/col blocks: 0:[0] (l=0), 1:[1..3], 2:[4..8], 3:[9..15].
  // D_l = Y(alpha) * X(beta) * Y(gamma);  Y is diag(cos) + antidiag(sin),
  // X[r][c] = sum_k (Vr Vr + Vi Vi)*cos(b w_k) + (Vi Vr - Vr Vi)*sin(b w_k).
  {
    const int i = t >> 4;
    const int j = t & 15;
    float d = 0.0f;
    const int bi = (i == 0) ? 0 : (i <= 3 ? 1 : (i <= 8 ? 2 : 3));
    const int bj = (j == 0) ? 0 : (j <= 3 ? 1 : (j <= 8 ? 2 : 3));
    if (bi == bj) {
      if (bi == 0) {
        d = 1.0f;                       // l=0: identity
      } else {
        const int l    = bi;
        const int base = (l == 1) ? 1 : (l == 2 ? 4 : 9);
        const int n    = 2 * l + 1;
        const int li = i - base, lj = j - base;
        const float* w  = (l == 1) ? K.w1  : (l == 2 ? K.w2  : K.w3);
        const float* Vr = (l == 1) ? K.Vr1 : (l == 2 ? K.Vr2 : K.Vr3);
        const float* Vi = (l == 1) ? K.Vi1 : (l == 2 ? K.Vi2 : K.Vi3);
        const int r0 = li, r1 = 2 * l - li;
        const int c0 = lj, c1 = 2 * l - lj;
        float x00 = 0.f, x01 = 0.f, x10 = 0.f, x11 = 0.f;
        for (int k = 0; k < n; ++k) {
          const float ck = __cosf(be * w[k]);
          const float sk = __sinf(be * w[k]);
          const float ar0 = Vr[r0 * n + k], ai0 = Vi[r0 * n + k];
          const float ar1 = Vr[r1 * n + k], ai1 = Vi[r1 * n + k];
          const float br0 = Vr[c0 * n + k], bi0 = Vi[c0 * n + k];
          const float br1 = Vr[c1 * n + k], bi1 = Vi[c1 * n + k];
          x00 += (ar0 * br0 + ai0 * bi0) * ck + (ai0 * br0 - ar0 * bi0) * sk;
          x01 += (ar0 * br1 + ai0 * bi1) * ck + (ai0 * br1 - ar0 * bi1) * sk;
          x10 += (ar1 * br0 + ai1 * bi0) * ck + (ai1 * br0 - ar1 * bi0) * sk;
          x11 += (ar1 * br1 + ai1 * bi1) * ck + (ai1 * br1 - ar1 * bi1) * sk;
        }
        // M = X * Y(gamma):  M[r][lj] = X[r][lj]*cos((lj-l)g) + X[r][2l-lj]*sin((l-lj)g)
        const float cG = __cosf((float)(lj - l) * ga);
        const float sG = __sinf((float)(l - lj) * ga);
        const float M0 = x00 * cG + x01 * sG;     // M[li][lj]
        const float M1 = x10 * cG + x11 * sG;     // M[2l-li][lj]
        // D = Y(alpha) * M:  D[li][lj] = cos((li-l)a)*M0 + sin((li-l)a)*M1
        const float cA = __cosf((float)(li - l) * al);
        const float sA = __sinf((float)(li - l) * al);
        d = cA * M0 + sA * M1;
      }
    }
    Dm[i][j] = d;
  }
  __syncthreads();

  // ---------------- Phase 2: out16x256 = Dfull @ x16x256 via WMMA -----------
  const int lane = t & 31;
  const int wv   = t >> 5;        // 8 waves, 2 column tiles each
  const int half = lane >> 4;     // 0: lanes 0-15, 1: lanes 16-31
  const int ln   = lane & 15;

  // A (16x4 f32) layout: lanes 0-15 -> M=lane, VGPR0=K0,VGPR1=K1;
  //                      lanes 16-31 -> M=lane-16, VGPR0=K2,VGPR1=K3.
  v2f a0, a1, a2, a3;
  a0.x = Dm[ln][0  + 2 * half]; a0.y = Dm[ln][1  + 2 * half];
  a1.x = Dm[ln][4  + 2 * half]; a1.y = Dm[ln][5  + 2 * half];
  a2.x = Dm[ln][8  + 2 * half]; a2.y = Dm[ln][9  + 2 * half];
  a3.x = Dm[ln][12 + 2 * half]; a3.y = Dm[ln][13 + 2 * half];

  const float* __restrict__ xb = x   + (size_t)b * 4096;
  float* __restrict__       ob = out + (size_t)b * 4096;

#pragma unroll
  for (int tt = 0; tt < 2; ++tt) {
    const int col = (wv * 2 + tt) * 16 + ln;
    // B (4x16 f32): VGPR0 holds rows K0 (lanes 0-15) / K2 (lanes 16-31), VGPR1 K1/K3.
    v2f b0, b1, b2, b3;
    b0.x = xb[(0  + 2 * half) * 256 + col]; b0.y = xb[(1  + 2 * half) * 256 + col];
    b1.x = xb[(4  + 2 * half) * 256 + col]; b1.y = xb[(5  + 2 * half) * 256 + col];
    b2.x = xb[(8  + 2 * half) * 256 + col]; b2.y = xb[(9  + 2 * half) * 256 + col];
    b3.x = xb[(12 + 2 * half) * 256 + col]; b3.y = xb[(13 + 2 * half) * 256 + col];

    v8f c = {};
    c = __builtin_amdgcn_wmma_f32_16x16x4_f32(false, a0, false, b0, (short)0, c, false, false);
    c = __builtin_amdgcn_wmma_f32_16x16x4_f32(false, a1, false, b1, (short)0, c, false, false);
    c = __builtin_amdgcn_wmma_f32_16x16x4_f32(false, a2, false, b2, (short)0, c, false, false);
    c = __builtin_amdgcn_wmma_f32_16x16x4_f32(false, a3, false, b3, (short)0, c, false, false);

    // C (16x16 f32): VGPR v holds row v (lanes 0-15) / row v+8 (lanes 16-31).
#pragma unroll
    for (int v = 0; v < 8; ++v)
      ob[(v + 8 * half) * 256 + col] = c[v];
  }
}

// ---------------------------------------------------------------------------
// Host: eigendecomposition constants of the x-axis generator in the real basis
// (deterministic, computed each launch; passed by value -> graph-capture safe).
// ---------------------------------------------------------------------------
namespace {
using cd = std::complex<double>;

void eig_herm(int n, cd A[7][7], double w[7], cd V[7][7]) {
  for (int i = 0; i < n; ++i)
    for (int j = 0; j < n; ++j) V[i][j] = (i == j) ? cd(1, 0) : cd(0, 0);
  for (int sweep = 0; sweep < 80; ++sweep) {
    double off = 0;
    for (int p = 0; p < n; ++p)
      for (int q = p + 1; q < n; ++q) off += std::norm(A[p][q]);
    if (off < 1e-24) break;
    for (int p = 0; p < n - 1; ++p)
      for (int q = p + 1; q < n; ++q) {
        const double apq = std::abs(A[p][q]);
        if (apq < 1e-15) continue;
        const cd ph = A[p][q] / apq;                      // e^{i phi}
        const double app = A[p][p].real(), aqq = A[q][q].real();
        const double th = (app - aqq) / (2.0 * apq);
        const double tt = (th >= 0 ? 1.0 : -1.0) / (std::fabs(th) + std::sqrt(1.0 + th * th));
        const double c = 1.0 / std::sqrt(1.0 + tt * tt);
        const double s = tt * c;
        const cd spc  = s * ph;                           // s e^{+i phi}
        const cd spcc = s * std::conj(ph);                // s e^{-i phi}
        for (int i = 0; i < n; ++i) {                     // A <- A U
          const cd aip = A[i][p], aiq = A[i][q];
          A[i][p] = c * aip + spcc * aiq;
          A[i][q] = -spc * aip + c * aiq;
        }
        for (int j = 0; j < n; ++j) {                     // A <- U^H A
          const cd apj = A[p][j], aqj = A[q][j];
          A[p][j] = c * apj + spc * aqj;
          A[q][j] = -spcc * apj + c * aqj;
        }
        for (int i = 0; i < n; ++i) {                     // V <- V U
          const cd vip = V[i][p], viq = V[i][q];
          V[i][p] = c * vip + spcc * viq;
          V[i][q] = -spc * vip + c * viq;
        }
      }
  }
  for (int k = 0; k < n; ++k) w[k] = A[k][k].real();
}

void build_l_consts(int l, float* wf, float* Vrf, float* Vif) {
  const int n = 2 * l + 1;
  double Jx[7][7] = {};
  for (int i = 0; i < n - 1; ++i) {
    const double m = -l + i;
    const double v = 0.5 * std::sqrt(l * (l + 1.0) - m * (m + 1.0));
    Jx[i + 1][i] += v;
    Jx[i][i + 1] += v;
  }
  cd U[7][7] = {};
  const double s = 1.0 / std::sqrt(2.0);
  U[l][l] = 1.0;
  for (int mu = 1; mu <= l; ++mu) {
    const double sgn = (mu & 1) ? -1.0 : 1.0;             // (-1)^mu
    U[l + mu][l - mu] = s;
    U[l + mu][l + mu] = s * sgn;
    U[l - mu][l - mu] = cd(0, s);
    U[l - mu][l + mu] = cd(0, -s * sgn);
  }
  cd T[7][7] = {}, Jr[7][7] = {};
  for (int i = 0; i < n; ++i)
    for (int k = 0; k < n; ++k)
      for (int j = 0; j < n; ++j) T[i][j] += U[i][k] * Jx[k][j];
  for (int i = 0; i < n; ++i)
    for (int j = 0; j < n; ++j)
      for (int k = 0; k < n; ++k) Jr[i][j] += T[i][k] * std::conj(U[j][k]);
  double w[7];
  cd V[7][7];
  eig_herm(n, Jr, w, V);
  for (int k = 0; k < n; ++k) wf[k] = (float)w[k];
  for (int i = 0; i < n; ++i)
    for (int k = 0; k < n; ++k) {
      Vrf[i * n + k] = (float)V[i][k].real();
      Vif[i * n + k] = (float)V[i][k].imag();
    }
}
}  // namespace

extern "C" void kernel_launch(void* const* d_in, const int* in_sizes, int n_in,
                              void* d_out, int out_size, void* d_ws, size_t ws_size,
                              hipStream_t stream) {
  (void)n_in; (void)out_size; (void)d_ws; (void)ws_size;
  RotConsts K;
  build_l_consts(1, K.w1, K.Vr1, K.Vi1);
  build_l_consts(2, K.w2, K.Vr2, K.Vi2);
  build_l_consts(3, K.w3, K.Vr3, K.Vi3);

  const float* gamma = (const float*)d_in[0];
  const float* beta  = (const float*)d_in[1];
  const float* alpha = (const float*)d_in[2];
  const float* x     = (const float*)d_in[3];
  float* out = (float*)d_out;

  const int B = in_sizes[0];   // 8192 batch elements, one block each
  rot_irreps_wmma<<<B, 256, 0, stream>>>(gamma, beta, alpha, x, out, K);
}